// ENDecoder_1778116461236
// MI455X (gfx1250) — compile-verified
//
#include <hip/hip_runtime.h>
#include <math.h>

#define NNODES 10000
#define NEDGES 160000
#define FEAT   128
#define NRBF   20
#define NCONV  3
#define CUT    5.0f

typedef float v2f __attribute__((ext_vector_type(2)));
typedef float v8f __attribute__((ext_vector_type(8)));

__device__ __forceinline__ v8f wmma_f32(v2f a, v2f b, v8f c) {
  // D = A(16x4 f32) * B(4x16 f32) + C(16x16 f32)
  return __builtin_amdgcn_wmma_f32_16x16x4_f32(false, a, false, b, (short)0, c,
                                               false, false);
}

// ---------------------------------------------------------------------------
// Edge geometry (layer invariant): unit vector, cosine envelope, rbf * env
// ---------------------------------------------------------------------------
__global__ __launch_bounds__(256) void edge_geom(
    const float* __restrict__ xyz, const int* __restrict__ nbr,
    float* __restrict__ unit, float* __restrict__ env_out,
    float* __restrict__ rbf_env) {
  int e = blockIdx.x * blockDim.x + threadIdx.x;
  if (e >= NEDGES) return;
  int i = nbr[2 * e], j = nbr[2 * e + 1];
  float dx = xyz[3 * j + 0] - xyz[3 * i + 0];
  float dy = xyz[3 * j + 1] - xyz[3 * i + 1];
  float dz = xyz[3 * j + 2] - xyz[3 * i + 2];
  float dist = sqrtf(dx * dx + dy * dy + dz * dz + 3.0e-15f);
  float inv = 1.0f / dist;
  unit[3 * e + 0] = dx * inv;
  unit[3 * e + 1] = dy * inv;
  unit[3 * e + 2] = dz * inv;
  float env = (dist < CUT)
                  ? 0.5f * (cosf(3.14159265358979323846f * dist / CUT) + 1.0f)
                  : 0.0f;
  env_out[e] = env;
  const float mu0 = 0.00673794699908547f;            // exp(-5)
  const float dmu = (1.0f - mu0) / (float)(NRBF - 1);
  const float q   = (2.0f / (float)NRBF) * (1.0f - mu0);
  const float beta = 1.0f / (q * q);
  float t = expf(-dist);
#pragma unroll
  for (int k = 0; k < NRBF; ++k) {
    float d = t - (mu0 + (float)k * dmu);
    rbf_env[(size_t)e * NRBF + k] = expf(-beta * d * d) * env;
  }
}

// ---------------------------------------------------------------------------
// Node GEMM with fp32 WMMA: C[N,NCOLS] = act(A[N,128] @ W[128,NCOLS] + bias)
// block = 256 threads (8 waves), 16 rows per block; wave w owns tiles
// n0 = 16*(w + 8*t), t = 0..TPW-1
// ---------------------------------------------------------------------------
template <int NCOLS, int TPW, bool SILU>
__global__ __launch_bounds__(256) void gemm_wmma(
    const float* __restrict__ A, const float* __restrict__ W,
    const float* __restrict__ bias, float* __restrict__ C) {
  __shared__ float shA[16 * 130];  // padded stride: avoid LDS bank conflicts
  const int tid = threadIdx.x;
  const int lane = tid & 31, wave = tid >> 5;
  const int nl = lane & 15, half = lane >> 4;
  const int rowbase = blockIdx.x * 16;

  for (int x = tid; x < 16 * FEAT; x += 256) {
    int r = x >> 7, c = x & 127;
    shA[r * 130 + c] = A[(size_t)(rowbase + r) * FEAT + c];
  }
  __syncthreads();

  v8f acc[TPW] = {};
#pragma unroll 8
  for (int k0 = 0; k0 < FEAT; k0 += 4) {
    const int kk = k0 + 2 * half;
    v2f a = *(const v2f*)&shA[nl * 130 + kk];
#pragma unroll
    for (int t = 0; t < TPW; ++t) {
      const int n = 16 * (wave + 8 * t) + nl;
      v2f b;
      b.x = W[(size_t)kk * NCOLS + n];
      b.y = W[(size_t)(kk + 1) * NCOLS + n];
      acc[t] = wmma_f32(a, b, acc[t]);
    }
  }

#pragma unroll
  for (int t = 0; t < TPW; ++t) {
    const int n = 16 * (wave + 8 * t) + nl;
    const float bv = bias[n];
#pragma unroll
    for (int r = 0; r < 8; ++r) {
      float val = acc[t][r] + bv;
      if (SILU) val = val / (1.0f + expf(-val));
      C[(size_t)(rowbase + r + 8 * half) * NCOLS + n] = val;
    }
  }
}

// ---------------------------------------------------------------------------
// Fused edge-message kernel. Block = 16 edges, 8 waves.
// WMMA: A = rbf_env (16 edges x 20), B = Wr (20 x 384).
// Wave w computes column tiles 16w (ds), 128+16w (gate_v), 256+16w (gate_u),
// so every lane holds ds/gv/gu for the SAME channel c and same 8 edges.
// Epilogue: prod = phi[j]*w_s, dv = gu*unit + gv*v[j]; atomic scatter to i.
// ---------------------------------------------------------------------------
__global__ __launch_bounds__(256) void edge_msg(
    const float* __restrict__ phi,      // [N,384]
    const float* __restrict__ rbf_env,  // [E,20]
    const float* __restrict__ env,      // [E]
    const float* __restrict__ unit,     // [E,3]
    const int* __restrict__ nbr,        // [E,2]
    const float* __restrict__ Wr,       // [20,384]
    const float* __restrict__ br,       // [384]
    const float* __restrict__ v_cur,    // [N,128,3]
    float* __restrict__ s_nxt,          // [N,128]
    float* __restrict__ v_nxt) {        // [N,128,3]
  __shared__ int shI[16], shJ[16];
  __shared__ float shEnv[16];
  __shared__ float shU[16][3];

  const int tid = threadIdx.x;
  const int lane = tid & 31, wave = tid >> 5;
  const int nl = lane & 15, half = lane >> 4;
  const int e0 = blockIdx.x * 16;

  if (tid < 16) {
    shI[tid] = nbr[2 * (e0 + tid)];
    shJ[tid] = nbr[2 * (e0 + tid) + 1];
    shEnv[tid] = env[e0 + tid];
  } else if (tid < 64) {
    int x = tid - 16;
    if (x < 48) shU[x / 3][x % 3] = unit[(size_t)e0 * 3 + x];
  }
  __syncthreads();

  const int c = 16 * wave + nl;  // feature channel 0..127
  v8f accS = {}, accV = {}, accU = {};
#pragma unroll
  for (int k0 = 0; k0 < NRBF; k0 += 4) {
    const int kk = k0 + 2 * half;
    v2f a = *(const v2f*)&rbf_env[(size_t)(e0 + nl) * NRBF + kk];
    v2f bS, bV, bU;
    bS.x = Wr[(size_t)kk * 384 + c];
    bS.y = Wr[(size_t)(kk + 1) * 384 + c];
    bV.x = Wr[(size_t)kk * 384 + 128 + c];
    bV.y = Wr[(size_t)(kk + 1) * 384 + 128 + c];
    bU.x = Wr[(size_t)kk * 384 + 256 + c];
    bU.y = Wr[(size_t)(kk + 1) * 384 + 256 + c];
    accS = wmma_f32(a, bS, accS);
    accV = wmma_f32(a, bV, accV);
    accU = wmma_f32(a, bU, accU);
  }

  const float brS = br[c], brV = br[128 + c], brU = br[256 + c];
#pragma unroll
  for (int r = 0; r < 8; ++r) {
    const int m = r + 8 * half;  // edge within block this acc element maps to
    const int jn = shJ[m], in_ = shI[m];
    const float ev = shEnv[m];
    const float* pj = &phi[(size_t)jn * 384];
    float ds = pj[c] * (accS[r] + ev * brS);
    float gv = pj[128 + c] * (accV[r] + ev * brV);
    float gu = pj[256 + c] * (accU[r] + ev * brU);
    atomicAdd(&s_nxt[(size_t)in_ * FEAT + c], ds);
    const float* vp = &v_cur[((size_t)jn * FEAT + c) * 3];
    float* vn = &v_nxt[((size_t)in_ * FEAT + c) * 3];
    atomicAdd(&vn[0], gu * shU[m][0] + gv * vp[0]);
    atomicAdd(&vn[1], gu * shU[m][1] + gv * vp[1]);
    atomicAdd(&vn[2], gu * shU[m][2] + gv * vp[2]);
  }
}

// ---------------------------------------------------------------------------
extern "C" void kernel_launch(void* const* d_in, const int* in_sizes, int n_in,
                              void* d_out, int out_size, void* d_ws,
                              size_t ws_size, hipStream_t stream) {
  const float* xyz  = (const float*)d_in[0];
  const int*   nbr  = (const int*)d_in[1];
  const float* cg_s = (const float*)d_in[2];
  const float* W1   = (const float*)d_in[3];
  const float* b1   = (const float*)d_in[4];
  const float* W2   = (const float*)d_in[5];
  const float* b2   = (const float*)d_in[6];
  const float* Wr   = (const float*)d_in[7];
  const float* br   = (const float*)d_in[8];

  float* ws  = (float*)d_ws;
  float* unit = ws;  ws += (size_t)NEDGES * 3;
  float* env  = ws;  ws += (size_t)NEDGES;
  float* rbfe = ws;  ws += (size_t)NEDGES * NRBF;
  float* H    = ws;  ws += (size_t)NNODES * FEAT;
  float* PHI  = ws;  ws += (size_t)NNODES * 3 * FEAT;
  float* sA   = ws;  ws += (size_t)NNODES * FEAT;
  float* sB   = ws;  ws += (size_t)NNODES * FEAT;
  float* vA   = ws;  ws += (size_t)NNODES * FEAT * 3;
  float* vB   = ws;  ws += (size_t)NNODES * FEAT * 3;

  edge_geom<<<(NEDGES + 255) / 256, 256, 0, stream>>>(xyz, nbr, unit, env, rbfe);
  hipMemcpyAsync(sA, cg_s, sizeof(float) * (size_t)NNODES * FEAT,
                 hipMemcpyDeviceToDevice, stream);
  hipMemsetAsync(vA, 0, sizeof(float) * (size_t)NNODES * FEAT * 3, stream);

  for (int l = 0; l < NCONV; ++l) {
    // next-state buffers start as current state; atomics accumulate deltas
    hipMemcpyAsync(sB, sA, sizeof(float) * (size_t)NNODES * FEAT,
                   hipMemcpyDeviceToDevice, stream);
    hipMemcpyAsync(vB, vA, sizeof(float) * (size_t)NNODES * FEAT * 3,
                   hipMemcpyDeviceToDevice, stream);
    gemm_wmma<FEAT, 1, true><<<NNODES / 16, 256, 0, stream>>>(
        sA, W1 + (size_t)l * FEAT * FEAT, b1 + (size_t)l * FEAT, H);
    gemm_wmma<3 * FEAT, 3, false><<<NNODES / 16, 256, 0, stream>>>(
        H, W2 + (size_t)l * FEAT * 3 * FEAT, b2 + (size_t)l * 3 * FEAT, PHI);
    edge_msg<<<NEDGES / 16, 256, 0, stream>>>(
        PHI, rbfe, env, unit, nbr, Wr + (size_t)l * NRBF * 3 * FEAT,
        br + (size_t)l * 3 * FEAT, vA, sB, vB);
    float* t;
    t = sA; sA = sB; sB = t;
    t = vA; vA = vB; vB = t;
  }

  hipMemcpyAsync(d_out, sA, sizeof(float) * (size_t)NNODES * FEAT,
                 hipMemcpyDeviceToDevice, stream);
  hipMemcpyAsync((float*)d_out + (size_t)NNODES * FEAT, vA,
                 sizeof(float) * (size_t)NNODES * FEAT * 3,
                 hipMemcpyDeviceToDevice, stream);
}